// SimpleAttention_31731218382924
// MI455X (gfx1250) — compile-verified
//
#include <hip/hip_runtime.h>
#include <hip/hip_bf16.h>
#include <math.h>

// ---------------------------------------------------------------------------
// SimpleAttention for MI455X (gfx1250): f32 WMMA 16x16x4 end-to-end.
// S=16, N=4096, H=32, D=128, M=8192.  Memory-bound (~470MB @ 23.3TB/s), so:
//  - every W / cache element is read from HBM exactly once (NT hints),
//  - grids are sized for HBM latency hiding (3072 / 1024 waves).
// ---------------------------------------------------------------------------

typedef __attribute__((ext_vector_type(2))) float v2f;
typedef __attribute__((ext_vector_type(8))) float v8f;

#define NMODEL 4096
#define SEQ    16
#define DHEAD  128
#define HHEADS 32
#define MCACHE 8192
#define KSEG   4      // k-split of the projection GEMMs (1024 rows each)
#define SPLITS 32     // softmax partial splits per head (4 blocks x 8 waves)

static __device__ inline v8f wmma_f32(v2f a, v2f b, v8f c) {
    // D = A(16x4 f32) x B(4x16 f32) + C(16x16 f32)
    return __builtin_amdgcn_wmma_f32_16x16x4_f32(
        /*neg_a=*/false, a, /*neg_b=*/false, b,
        /*c_mod=*/(short)0, c, /*reuse_a=*/false, /*reuse_b=*/false);
}

// ---------------------------------------------------------------------------
// Kernel 1: partial Q/K/V projections, k-split for parallelism.
// grid = 3 mats * KSEG * 32 tile-groups = 384 blocks x 256 threads (8 waves).
// Block owns (matrix, k-range of 1024, 8 n-tiles); wave w owns n-tile
// n0=(tg*8+w)*16.  X k-chunk staged in LDS (stride 516: 16B aligned,
// conflict-free).  Each W element is streamed from HBM exactly once (NT).
// ---------------------------------------------------------------------------
__global__ __launch_bounds__(256)
void qkv_proj_kernel(const float* __restrict__ X,
                     const float* __restrict__ Wq,
                     const float* __restrict__ Wk,
                     const float* __restrict__ Wv,
                     float* __restrict__ Part) {   // [3*KSEG][16][4096]
    __shared__ float xs[SEQ * 516];

    const int tid  = threadIdx.x;
    const int wv   = tid >> 5;
    const int l    = tid & 31;
    const int ln   = l & 15;
    const int half = l >> 4;

    const int mat   = blockIdx.x >> 7;          // 0..2
    const int rem   = blockIdx.x & 127;
    const int kseg  = rem >> 5;                 // 0..3
    const int tg    = rem & 31;                 // 0..31
    const int n0    = (tg * 8 + wv) * 16;
    const int kbase = kseg * 1024;

    const float* W = (mat == 0) ? Wq : (mat == 1) ? Wk : Wv;

    v8f c = {};

    for (int chunk = 0; chunk < 1024; chunk += 512) {
        __syncthreads();
        // cooperative load X[:, kbase+chunk : +512] -> xs (float4, coalesced)
        for (int i = tid * 4; i < SEQ * 512; i += 256 * 4) {
            const int row = i >> 9;
            const int col = i & 511;
            const float4 xv = *(const float4*)(X + row * NMODEL + kbase + chunk + col);
            *(float4*)(xs + row * 516 + col) = xv;
        }
        __syncthreads();

        for (int kk = 0; kk < 512; kk += 4) {
            // A 16x4: lane ln holds row ln, K = 2*half, 2*half+1
            const v2f a = *(const v2f*)(xs + ln * 516 + kk + 2 * half);
            // B 4x16: lane holds col n0+ln, rows k = 2*half, 2*half+1 (NT stream)
            const int rb = (kbase + chunk + kk + 2 * half) * NMODEL + n0 + ln;
            const v2f b = { __builtin_nontemporal_load(W + rb),
                            __builtin_nontemporal_load(W + rb + NMODEL) };
            c = wmma_f32(a, b, c);
        }
    }

    // C layout: VGPR r -> row (r + 8*half), lane -> col n0+ln
    float* dst = Part + (mat * KSEG + kseg) * SEQ * NMODEL;
#pragma unroll
    for (int r = 0; r < 8; ++r) {
        const int s = r + 8 * half;
        dst[s * NMODEL + n0 + ln] = c[r];
    }
}

// ---------------------------------------------------------------------------
// Kernel 1b: reduce the KSEG partial sums into Q/K/V (contiguous, 3*16*4096).
// ---------------------------------------------------------------------------
__global__ __launch_bounds__(256)
void qkv_reduce_kernel(const float* __restrict__ Part,  // [3*KSEG][65536]
                       float* __restrict__ QKV) {       // [3][65536]
    const int stride = gridDim.x * 256 * 4;
    for (int base = (blockIdx.x * 256 + threadIdx.x) * 4;
         base < 3 * SEQ * NMODEL; base += stride) {
        const int mat = base >> 16;          // /65536
        const int idx = base & 65535;
        float4 s = *(const float4*)(Part + (mat * KSEG + 0) * SEQ * NMODEL + idx);
#pragma unroll
        for (int k = 1; k < KSEG; ++k) {
            const float4 p = *(const float4*)(Part + (mat * KSEG + k) * SEQ * NMODEL + idx);
            s.x += p.x; s.y += p.y; s.z += p.z; s.w += p.w;
        }
        *(float4*)(QKV + base) = s;
    }
}

// ---------------------------------------------------------------------------
// Kernel 2: flash-style attention partials.
// grid = 32 heads * 4 segs = 128 blocks x 256 threads (8 waves).
// Split id = seg*8 + wave covers cache rows [split*256, split*256+256).
// Rows [P, P+16) substituted with projected K/V via per-lane pointer select
// (no divergent branch -> EXEC all-ones at every WMMA).
// ---------------------------------------------------------------------------
__global__ __launch_bounds__(256)
void attn_partial_kernel(const float* __restrict__ Q,
                         const float* __restrict__ Knew,
                         const float* __restrict__ Vnew,
                         const float* __restrict__ cacheK,
                         const float* __restrict__ cacheV,
                         const int*   __restrict__ Pptr,
                         float* __restrict__ Opart,   // [H][SPLITS][16][128]
                         float* __restrict__ Mstat,   // [H][SPLITS][16]
                         float* __restrict__ Lstat) { // [H][SPLITS][16]
    const int h     = blockIdx.x >> 2;
    const int seg   = blockIdx.x & 3;
    const int tid   = threadIdx.x;
    const int wv    = tid >> 5;
    const int l     = tid & 31;
    const int ln    = l & 15;
    const int half  = l >> 4;
    const int split = seg * 8 + wv;          // 0..31
    const int P     = *Pptr;

    __shared__ float qs[SEQ * 130];      // q tile, padded stride
    __shared__ float ps[8][SEQ * 18];    // per-wave p tile, padded stride

    // stage q for this head: Q[s*4096 + h*128 + d] -> qs[s*130 + d]
    for (int i = tid; i < SEQ * DHEAD; i += 256) {
        const int s = i >> 7;
        const int d = i & 127;
        qs[s * 130 + d] = Q[s * NMODEL + h * DHEAD + d];
    }
    __syncthreads();

    const float* Kbase = cacheK + (size_t)h * MCACHE * DHEAD;
    const float* Vbase = cacheV + (size_t)h * MCACHE * DHEAD;
    const float* Ksub  = Knew + h * DHEAD;   // row s at s*NMODEL
    const float* Vsub  = Vnew + h * DHEAD;

    v8f acc[8];
#pragma unroll
    for (int dc = 0; dc < 8; ++dc) acc[dc] = (v8f){};
    float mrow[8], lrow[8];
#pragma unroll
    for (int r = 0; r < 8; ++r) { mrow[r] = -__builtin_inff(); lrow[r] = 0.0f; }

    for (int t = 0; t < 16; ++t) {
        const int m0 = split * 256 + t * 16;

        // ---- scores = q (16x128) . K_tile^T  -> 16x16 ----
        const int mk = m0 + ln;                       // this lane's cache row
        const float* krow = (mk >= P && mk < P + SEQ)
                          ? (Ksub + (mk - P) * NMODEL)
                          : (Kbase + (size_t)mk * DHEAD);
        v8f sc = {};
#pragma unroll
        for (int d0 = 0; d0 < DHEAD; d0 += 4) {
            const v2f a = *(const v2f*)(qs + ln * 130 + d0 + 2 * half);
            const v2f b = __builtin_nontemporal_load(
                              (const v2f*)(krow + d0 + 2 * half));
            sc = wmma_f32(a, b, sc);
        }

        // ---- online softmax update (rows r+8*half, cols across 16 lanes) ----
#pragma unroll
        for (int r = 0; r < 8; ++r) {
            float mx = sc[r];
            mx = fmaxf(mx, __shfl_xor(mx, 1, 32));
            mx = fmaxf(mx, __shfl_xor(mx, 2, 32));
            mx = fmaxf(mx, __shfl_xor(mx, 4, 32));
            mx = fmaxf(mx, __shfl_xor(mx, 8, 32));
            const float mnew  = fmaxf(mrow[r], mx);
            const float scale = __expf(mrow[r] - mnew);
            const float p     = __expf(sc[r] - mnew);
            float srow = p;
            srow += __shfl_xor(srow, 1, 32);
            srow += __shfl_xor(srow, 2, 32);
            srow += __shfl_xor(srow, 4, 32);
            srow += __shfl_xor(srow, 8, 32);
            lrow[r] = lrow[r] * scale + srow;
            mrow[r] = mnew;
#pragma unroll
            for (int dc = 0; dc < 8; ++dc) acc[dc][r] *= scale;
            // p tile -> LDS in [s][m] order for A-layout reload
            ps[wv][(r + 8 * half) * 18 + ln] = p;
        }
        __builtin_amdgcn_wave_barrier();   // keep LDS store->load ordered

        // ---- out += p (16x16) . V_tile (16x128) ----
#pragma unroll
        for (int k0 = 0; k0 < 16; k0 += 4) {
            const int r0 = m0 + k0 + 2 * half;
            const int r1 = r0 + 1;
            const float* vrow0 = (r0 >= P && r0 < P + SEQ)
                               ? (Vsub + (r0 - P) * NMODEL)
                               : (Vbase + (size_t)r0 * DHEAD);
            const float* vrow1 = (r1 >= P && r1 < P + SEQ)
                               ? (Vsub + (r1 - P) * NMODEL)
                               : (Vbase + (size_t)r1 * DHEAD);
            const v2f a = *(const v2f*)(&ps[wv][ln * 18 + k0 + 2 * half]);
#pragma unroll
            for (int dc = 0; dc < 8; ++dc) {
                const v2f b = { __builtin_nontemporal_load(vrow0 + dc * 16 + ln),
                                __builtin_nontemporal_load(vrow1 + dc * 16 + ln) };
                acc[dc] = wmma_f32(a, b, acc[dc]);
            }
        }
    }

    // ---- write partials ----
    const int pbase = (h * SPLITS + split) * SEQ * DHEAD;
#pragma unroll
    for (int dc = 0; dc < 8; ++dc)
#pragma unroll
        for (int r = 0; r < 8; ++r) {
            const int s = r + 8 * half;
            Opart[pbase + s * DHEAD + dc * 16 + ln] = acc[dc][r];
        }
    if (ln == 0) {
        const int sbase = (h * SPLITS + split) * SEQ;
#pragma unroll
        for (int r = 0; r < 8; ++r) {
            Mstat[sbase + r + 8 * half] = mrow[r];
            Lstat[sbase + r + 8 * half] = lrow[r];
        }
    }
}

// ---------------------------------------------------------------------------
// Kernel 3: deterministic log-sum-exp merge of SPLITS partials per head.
// Output flat layout matches reference reshape: (h,s,d) -> h*S*D + s*D + d.
// ---------------------------------------------------------------------------
__global__ __launch_bounds__(256)
void attn_merge_kernel(const float* __restrict__ Opart,
                       const float* __restrict__ Mstat,
                       const float* __restrict__ Lstat,
                       float* __restrict__ out) {
    const int h = blockIdx.x;
    for (int i = threadIdx.x; i < SEQ * DHEAD; i += 256) {
        const int s = i >> 7;
        const int d = i & 127;
        float M = -__builtin_inff();
#pragma unroll
        for (int w = 0; w < SPLITS; ++w)
            M = fmaxf(M, Mstat[(h * SPLITS + w) * SEQ + s]);
        float L = 0.0f, o = 0.0f;
#pragma unroll 8
        for (int w = 0; w < SPLITS; ++w) {
            const float e = __expf(Mstat[(h * SPLITS + w) * SEQ + s] - M);
            L += Lstat[(h * SPLITS + w) * SEQ + s] * e;
            o += __builtin_nontemporal_load(
                     Opart + ((h * SPLITS + w) * SEQ + s) * DHEAD + d) * e;
        }
        out[h * SEQ * DHEAD + s * DHEAD + d] = o / L;
    }
}

// ---------------------------------------------------------------------------
extern "C" void kernel_launch(void* const* d_in, const int* in_sizes, int n_in,
                              void* d_out, int out_size, void* d_ws, size_t ws_size,
                              hipStream_t stream) {
    const float* X      = (const float*)d_in[0];
    const float* Wq     = (const float*)d_in[1];
    const float* Wk     = (const float*)d_in[2];
    const float* Wv     = (const float*)d_in[3];
    const float* cacheK = (const float*)d_in[4];
    const float* cacheV = (const float*)d_in[5];
    const int*   Pptr   = (const int*)d_in[6];
    float* out = (float*)d_out;

    // workspace layout (floats)
    float* ws    = (float*)d_ws;
    float* QKV   = ws;                                   // 3*16*4096
    float* Q     = QKV;
    float* K     = QKV + SEQ * NMODEL;
    float* V     = QKV + 2 * SEQ * NMODEL;
    float* Part  = QKV + 3 * SEQ * NMODEL;               // 3*KSEG*16*4096
    float* Opart = Part + 3 * KSEG * SEQ * NMODEL;       // 32*SPLITS*16*128
    float* Mstat = Opart + HHEADS * SPLITS * SEQ * DHEAD;// 32*SPLITS*16
    float* Lstat = Mstat + HHEADS * SPLITS * SEQ;        // 32*SPLITS*16

    qkv_proj_kernel<<<3 * KSEG * 32, 256, 0, stream>>>(X, Wq, Wk, Wv, Part);
    qkv_reduce_kernel<<<96, 256, 0, stream>>>(Part, QKV);
    attn_partial_kernel<<<HHEADS * 4, 256, 0, stream>>>(Q, K, V, cacheK, cacheV,
                                                        Pptr, Opart, Mstat, Lstat);
    attn_merge_kernel<<<HHEADS, 256, 0, stream>>>(Opart, Mstat, Lstat, out);
}